// TensorOperatorLayer_45543833206791
// MI455X (gfx1250) — compile-verified
//
#include <hip/hip_runtime.h>
#include <stdint.h>

// Problem dims (fixed by the reference)
#define S_TOT   16384
#define F_DIM   32
#define D_DIM   16
#define B_DIM   4
#define C_DIM   10
#define NMID    14

typedef float v2f __attribute__((ext_vector_type(2)));
typedef float v8f __attribute__((ext_vector_type(8)));
typedef unsigned int v4u __attribute__((ext_vector_type(4)));
typedef int  v8i __attribute__((ext_vector_type(8)));
typedef int  v4i __attribute__((ext_vector_type(4)));

// LDS layout (all sizes in floats). Strides padded for conflict-free access.
#define XO_S    516                        // xo[s][b*4+v][u], s-stride (512+4)
#define XO_SZ   (16*XO_S)                  // 8256
#define ST_C    68                         // state c-row stride (64+4)
#define ST_S    684                        // state sample stride (10*68+4)
#define ST_SZ   (16*ST_S)                  // 10944
#define W_S     260                        // W[s][r][q], s-stride (256+4)
#define W_SZ    (16*W_S)                   // 4160
#define WAVE_FLOATS (XO_SZ + 2*ST_SZ + W_SZ)      // 34304
#define LDSA_FLOATS (D_DIM*F_DIM*D_DIM)           // 8192 (one A_mid core)
#define SMEM_FLOATS (LDSA_FLOATS + 2*WAVE_FLOATS) // 76800
#define SMEM_BYTES  (SMEM_FLOATS*4)               // 307200 (< 320KB WGP LDS)

static __device__ inline v8f wmma4(v2f a, v2f b, v8f c) {
  // D = A(16x4 f32) * B(4x16 f32) + C(16x16 f32), exact fp32 path
  return __builtin_amdgcn_wmma_f32_16x16x4_f32(false, a, false, b, (short)0, c,
                                               false, false);
}

__global__ void __launch_bounds__(64)
tensor_mpo_kernel(const float* __restrict__ x,
                  const float* __restrict__ op,     // (B,F,F,B)
                  const float* __restrict__ A0,     // (C,F,D)
                  const float* __restrict__ Amid,   // (NMID,D,F,D)
                  const float* __restrict__ Alast,  // (D,F)
                  float* __restrict__ out)          // (S,C)
{
  extern __shared__ float smem[];
  float* ldsA = smem;                              // staged A_mid core (shared)

  const int tid  = threadIdx.x;
  const int lane = tid & 31;
  const int wv   = tid >> 5;
  const int h    = lane >> 4;                      // lane half (0/1)
  const int m    = lane & 15;                      // lane % 16

  float* wbase = smem + LDSA_FLOATS + wv * WAVE_FLOATS;
  float* xo  = wbase;                              // xo[s][bv][u]
  float* stA = wbase + XO_SZ;                      // state buffers
  float* stB = stA + ST_SZ;
  float* Wb  = stB + ST_SZ;                        // W / y0 / Z scratch
  float* y0  = Wb;                                 // y0[s][v][u], stride 132

  const int s0 = (blockIdx.x * 2 + wv) * 16;       // 16-sample tile base

  const v8f vz = {0.f,0.f,0.f,0.f,0.f,0.f,0.f,0.f};

  // ---- x A-fragments (M=s rows, K=d), kept in registers all kernel ----
  v2f xf[8];
  #pragma unroll
  for (int k = 0; k < 8; ++k)
    xf[k] = *(const v2f*)(x + (size_t)(s0 + m) * F_DIM + 4*k + 2*h);

  // ---- xo[s,b,u,v] = sum_d x[s,d] * op[b,u,d,v]  (M=s,K=d,N=(b,v,u)) ----
  for (int jn = 0; jn < 32; ++jn) {
    const int bv = jn >> 1, uh = jn & 1;
    const int b = bv >> 2, v = bv & 3;
    const int u = uh*16 + m;
    v8f c = vz;
    #pragma unroll
    for (int k = 0; k < 8; ++k) {
      const int d = 4*k + h;
      v2f bf = { op[((b*F_DIM + u)*F_DIM + d    )*B_DIM + v],
                 op[((b*F_DIM + u)*F_DIM + d + 2)*B_DIM + v] };
      c = wmma4(xf[k], bf, c);
    }
    #pragma unroll
    for (int i = 0; i < 8; ++i)
      xo[(i + 8*h)*XO_S + bv*32 + u] = c[i];
  }

  // ---- y0[s,u,v] = sum_d x[s,d] * op[0,u,d,v] ----
  for (int jn = 0; jn < 8; ++jn) {
    const int v = jn >> 1, uh = jn & 1;
    const int u = uh*16 + m;
    v8f c = vz;
    #pragma unroll
    for (int k = 0; k < 8; ++k) {
      const int d = 4*k + h;
      v2f bf = { op[(u*F_DIM + d    )*B_DIM + v],
                 op[(u*F_DIM + d + 2)*B_DIM + v] };
      c = wmma4(xf[k], bf, c);
    }
    #pragma unroll
    for (int i = 0; i < 8; ++i)
      y0[(i + 8*h)*132 + v*32 + u] = c[i];
  }

  // ---- st0[s,c,v,q] = sum_u y0[s,u,v] * A0[c,u,q] ----
  for (int v = 0; v < 4; ++v) {
    v2f yf[8];
    #pragma unroll
    for (int k = 0; k < 8; ++k)
      yf[k] = *(const v2f*)(y0 + m*132 + v*32 + 4*k + 2*h);
    for (int cc = 0; cc < C_DIM; ++cc) {
      v8f acc = vz;
      #pragma unroll
      for (int k = 0; k < 8; ++k) {
        const int u = 4*k + h;
        v2f bf = { A0[(cc*F_DIM + u    )*D_DIM + m],
                   A0[(cc*F_DIM + u + 2)*D_DIM + m] };
        acc = wmma4(yf[k], bf, acc);
      }
      #pragma unroll
      for (int i = 0; i < 8; ++i)
        stA[(i + 8*h)*ST_S + cc*ST_C + v*16 + m] = acc[i];
    }
  }

  float* cur = stA;
  float* nxt = stB;

  // ================= scan over 14 middle carriages =================
  for (int t = 0; t < NMID; ++t) {
    __syncthreads();  // everyone done reading previous ldsA
#if __has_builtin(__builtin_amdgcn_tensor_load_to_lds)
    if (wv == 0) {
      // TDM: DMA this step's A core (contiguous 32KB) into LDS
      const uint64_t ga = (uint64_t)(uintptr_t)(Amid + (size_t)t * LDSA_FLOATS);
      v4u g0;
      g0[0] = 1u;                                     // count=1, user desc
      g0[1] = (uint32_t)(uintptr_t)ldsA;              // lds_addr (bytes)
      g0[2] = (uint32_t)ga;                           // global_addr[95:64]
      g0[3] = ((uint32_t)(ga >> 32) & 0x01FFFFFFu) | (2u << 30); // addr hi | type=2
      v8i g1;
      g1[0] = (int)(2u << 16);                        // data_size=4B, wg_mask=0
      g1[1] = (int)((LDSA_FLOATS & 0xFFFFu) << 16);   // tensor_dim0 low16
      g1[2] = (int)((LDSA_FLOATS >> 16) | (1u << 16));// dim0 hi16 | tensor_dim1=1
      g1[3] = (int)((LDSA_FLOATS & 0xFFFFu) << 16);   // tile_dim0 = 8192
      g1[4] = 0;                                      // tile_dim1/2 unused
      g1[5] = (int)LDSA_FLOATS;                       // tensor_dim0_stride
      g1[6] = 0; g1[7] = 0;
      v4i gz4 = {0,0,0,0};
      v8i gz8 = {0,0,0,0,0,0,0,0};
      __builtin_amdgcn_tensor_load_to_lds(g0, g1, gz4, gz4, gz8, 0);
      __builtin_amdgcn_s_wait_tensorcnt((short)0);
    }
#else
    for (int idx = tid; idx < LDSA_FLOATS; idx += 64)
      ldsA[idx] = Amid[(size_t)t * LDSA_FLOATS + idx];
#endif
    __syncthreads();  // A core visible to both waves

    for (int bv = 0; bv < 16; ++bv) {
      const int b = bv >> 2, v = bv & 3;

      // --- GEMM1: W[s,r,q] = sum_u xo[s,b,u,v] * A_t[r,u,q] ---
      for (int rb = 0; rb < 2; ++rb) {
        v8f cf[8];
        #pragma unroll
        for (int rr = 0; rr < 8; ++rr) cf[rr] = vz;
        #pragma unroll
        for (int k = 0; k < 8; ++k) {
          v2f a = *(const v2f*)(xo + m*XO_S + bv*32 + 4*k + 2*h);
          #pragma unroll
          for (int rr = 0; rr < 8; ++rr) {
            const int r = rb*8 + rr;
            const int u = 4*k + h;
            v2f bf = { ldsA[(r*F_DIM + u    )*D_DIM + m],
                       ldsA[(r*F_DIM + u + 2)*D_DIM + m] };
            cf[rr] = wmma4(a, bf, cf[rr]);
          }
        }
        #pragma unroll
        for (int rr = 0; rr < 8; ++rr) {
          #pragma unroll
          for (int i = 0; i < 8; ++i)
            Wb[(i + 8*h)*W_S + (rb*8 + rr)*16 + m] = cf[rr][i];
        }
      }

      // --- GEMM2 (per sample): st'[s,c,v,:] += st[s,c,b,:] * W[s] ---
      for (int s = 0; s < 16; ++s) {
        v2f af[4];
        #pragma unroll
        for (int k = 0; k < 4; ++k)
          af[k] = *(const v2f*)(cur + s*ST_S + m*ST_C + b*16 + 4*k + 2*h);
        v8f acc;
        if (b == 0) {
          acc = vz;
        } else {
          #pragma unroll
          for (int i = 0; i < 8; ++i)
            acc[i] = nxt[s*ST_S + (i + 8*h)*ST_C + v*16 + m];
        }
        #pragma unroll
        for (int k = 0; k < 4; ++k) {
          const int r = 4*k + h;
          v2f bf = { Wb[s*W_S + r*16 + m],
                     Wb[s*W_S + (r + 2)*16 + m] };
          acc = wmma4(af[k], bf, acc);
        }
        #pragma unroll
        for (int i = 0; i < 8; ++i)
          if (h == 0 || i < 2)   // only rows c<10 are real state
            nxt[s*ST_S + (i + 8*h)*ST_C + v*16 + m] = acc[i];
      }
    }
    float* tp = cur; cur = nxt; nxt = tp;
  }

  // ================= final carriage =================
  // z[s,b,u] = sum_d x[s,d]*op[b,u,d,B-1]   (into xo region, stride 132)
  for (int jn = 0; jn < 8; ++jn) {
    const int b = jn >> 1, uh = jn & 1;
    const int u = uh*16 + m;
    v8f acc = vz;
    #pragma unroll
    for (int k = 0; k < 8; ++k) {
      const int d = 4*k + h;
      v2f bf = { op[((b*F_DIM + u)*F_DIM + d    )*B_DIM + (B_DIM-1)],
                 op[((b*F_DIM + u)*F_DIM + d + 2)*B_DIM + (B_DIM-1)] };
      acc = wmma4(xf[k], bf, acc);
    }
    #pragma unroll
    for (int i = 0; i < 8; ++i)
      xo[(i + 8*h)*132 + b*32 + u] = acc[i];
  }
  // Z[s,b,r] = sum_u z[s,b,u]*Alast[r,u]    (into Wb region, stride 68)
  for (int b = 0; b < 4; ++b) {
    v2f zf[8];
    #pragma unroll
    for (int k = 0; k < 8; ++k)
      zf[k] = *(const v2f*)(xo + m*132 + b*32 + 4*k + 2*h);
    v8f acc = vz;
    #pragma unroll
    for (int k = 0; k < 8; ++k) {
      const int u = 4*k + h;
      v2f bf = { Alast[m*F_DIM + u], Alast[m*F_DIM + u + 2] };
      acc = wmma4(zf[k], bf, acc);
    }
    #pragma unroll
    for (int i = 0; i < 8; ++i)
      Wb[(i + 8*h)*ST_C + b*16 + m] = acc[i];
  }
  // out[s,c] = sum_j st[s,c,j] * Z[s,j]
  for (int o = lane; o < 16 * C_DIM; o += 32) {
    const int s = o / C_DIM, cc = o % C_DIM;
    float acc = 0.f;
    const float* sp = cur + s*ST_S + cc*ST_C;
    const float* zp = Wb + s*ST_C;
    #pragma unroll
    for (int j = 0; j < 64; ++j) acc += sp[j] * zp[j];
    out[(size_t)(s0 + s) * C_DIM + cc] = acc;
  }
}

extern "C" void kernel_launch(void* const* d_in, const int* in_sizes, int n_in,
                              void* d_out, int out_size, void* d_ws, size_t ws_size,
                              hipStream_t stream) {
  const float* x     = (const float*)d_in[0];
  const float* op    = (const float*)d_in[1];
  const float* A0    = (const float*)d_in[2];
  const float* Amid  = (const float*)d_in[3];
  const float* Alast = (const float*)d_in[4];
  float* out = (float*)d_out;
  (void)in_sizes; (void)n_in; (void)out_size; (void)d_ws; (void)ws_size;

  dim3 grid(S_TOT / 32);   // 2 waves per WG, 16 samples per wave
  dim3 block(64);
  hipLaunchKernelGGL(tensor_mpo_kernel, grid, block, SMEM_BYTES, stream,
                     x, op, A0, Amid, Alast, out);
}